// AdditiveAttention_1683627180788
// MI455X (gfx1250) — compile-verified
//
#include <hip/hip_runtime.h>

typedef __attribute__((ext_vector_type(2))) float v2f;
typedef __attribute__((ext_vector_type(8))) float v8f;

#define NEG_INF_F (-1e6f)

// tanh(x) = 1 - 2/(e^{2x}+1); e^{2x} = exp2(x * 2*log2(e))
// Uses the hardware trans pipe: v_exp_f32 + v_rcp_f32.
__device__ __forceinline__ float fast_tanh(float x) {
    float e = __builtin_amdgcn_exp2f(x * 2.8853900817779268f);
    return 1.0f - 2.0f * __builtin_amdgcn_rcpf(e + 1.0f);
}

// ---------------------------------------------------------------------------
// Generic batched fp32 GEMM using V_WMMA_F32_16X16X4_F32.
//   C[b] (MxN, row-major, ldc=N) = A[b] (MxK, row-major, lda=K) * B[b] (KxN, ldb=N)
// One 16x16 output tile per wave. All dims must divide evenly by 16/4 and the
// grid must supply exactly batches*tilesPerBatch waves (keeps EXEC all-ones).
// ---------------------------------------------------------------------------
__global__ void wmma_gemm_f32(const float* __restrict__ A,
                              const float* __restrict__ B,
                              float* __restrict__ C,
                              int K, int N,
                              int tilesPerBatch, int tilesN,
                              long strideA, long strideB, long strideC) {
    const int wavesPerBlock = blockDim.x >> 5;
    const int waveId = blockIdx.x * wavesPerBlock + (threadIdx.x >> 5);
    const int batch = waveId / tilesPerBatch;
    const int t     = waveId % tilesPerBatch;
    const int tileM = t / tilesN;
    const int tileN = t % tilesN;

    const int lane = threadIdx.x & 31;
    const int half = lane >> 4;   // 0: lanes 0-15, 1: lanes 16-31
    const int l16  = lane & 15;

    const float* a_row = A + (long)batch * strideA + (long)(tileM * 16 + l16) * K;
    const float* b_col = B + (long)batch * strideB + (tileN * 16 + l16);

    v8f c = {};
    for (int k0 = 0; k0 < K; k0 += 4) {
        const int ka = k0 + 2 * half;           // lane half selects K {0,1} vs {2,3}
        v2f a, b;
        a.x = a_row[ka];
        a.y = a_row[ka + 1];
        b.x = b_col[(long)ka * N];
        b.y = b_col[(long)(ka + 1) * N];
        c = __builtin_amdgcn_wmma_f32_16x16x4_f32(
                /*neg_a=*/false, a, /*neg_b=*/false, b,
                /*c_mod=*/(short)0, c, /*reuse_a=*/false, /*reuse_b=*/false);
    }

    // C/D layout: VGPR j holds row (j + 8*half), col = l16
    float* c_out = C + (long)batch * strideC
                 + (long)(tileM * 16 + 8 * half) * N + tileN * 16 + l16;
#pragma unroll
    for (int j = 0; j < 8; ++j)
        c_out[(long)j * N] = c[j];
}

// ---------------------------------------------------------------------------
// Additive-attention scores + masked softmax.
// One block per (b,q) row; 256 threads; each thread owns 4 of the K=1024 keys.
//   s[k] = sum_h w_v[h] * tanh(qproj[b,q,h] + kproj[b,k,h]),  masked -> -1e6
//   probs[b,q,:] = softmax(s)
// ---------------------------------------------------------------------------
__global__ void score_softmax_kernel(const float* __restrict__ qproj,   // [B*Q,128]
                                     const float* __restrict__ kproj,   // [B*K,128]
                                     const float* __restrict__ wv_g,    // [128]
                                     const int*   __restrict__ valid_lens, // [B]
                                     float* __restrict__ probs) {       // [B*Q,1024]
    const int bq  = blockIdx.x;        // 0..1023
    const int b   = bq >> 6;           // Q = 64
    const int tid = threadIdx.x;       // 0..255

    __shared__ float qp[128];
    __shared__ float wv[128];
    __shared__ float red[256];

    if (tid < 128) {
        qp[tid] = qproj[(size_t)bq * 128 + tid];
        wv[tid] = wv_g[tid];
    }
    __syncthreads();

    const int vl = valid_lens[b];

    float s[4];
#pragma unroll
    for (int i = 0; i < 4; ++i) {
        const int k = tid + i * 256;
        const float4* kr = (const float4*)(kproj + ((size_t)(b * 1024 + k)) * 128);
        const float4* qv4 = (const float4*)qp;
        const float4* wv4 = (const float4*)wv;
        float acc = 0.0f;
#pragma unroll 8
        for (int h4 = 0; h4 < 32; ++h4) {
            float4 kv = kr[h4];
            float4 qv = qv4[h4];
            float4 wt = wv4[h4];
            acc += wt.x * fast_tanh(qv.x + kv.x);
            acc += wt.y * fast_tanh(qv.y + kv.y);
            acc += wt.z * fast_tanh(qv.z + kv.z);
            acc += wt.w * fast_tanh(qv.w + kv.w);
        }
        s[i] = (k < vl) ? acc : NEG_INF_F;
    }

    // --- block max reduction over 1024 scores ---
    float m = fmaxf(fmaxf(s[0], s[1]), fmaxf(s[2], s[3]));
    red[tid] = m;
    __syncthreads();
    for (int off = 128; off > 0; off >>= 1) {
        if (tid < off) red[tid] = fmaxf(red[tid], red[tid + off]);
        __syncthreads();
    }
    m = red[0];
    __syncthreads();

    // --- exp + block sum reduction ---
    float e[4];
    float lsum = 0.0f;
#pragma unroll
    for (int i = 0; i < 4; ++i) {
        e[i] = __builtin_amdgcn_exp2f((s[i] - m) * 1.4426950408889634f);
        lsum += e[i];
    }
    red[tid] = lsum;
    __syncthreads();
    for (int off = 128; off > 0; off >>= 1) {
        if (tid < off) red[tid] += red[tid + off];
        __syncthreads();
    }
    const float inv = __builtin_amdgcn_rcpf(red[0]);

#pragma unroll
    for (int i = 0; i < 4; ++i)
        probs[(size_t)bq * 1024 + tid + i * 256] = e[i] * inv;
}

// ---------------------------------------------------------------------------
// Launch: proj(q) -> proj(k) -> score+softmax -> PV GEMM, all on `stream`.
// Workspace (floats): qproj[131072] | kproj[2097152] | probs[1048576] = 12.5 MB
// (everything stays resident in the 192 MB L2).
// ---------------------------------------------------------------------------
extern "C" void kernel_launch(void* const* d_in, const int* in_sizes, int n_in,
                              void* d_out, int out_size, void* d_ws, size_t ws_size,
                              hipStream_t stream) {
    const float* queries = (const float*)d_in[0];  // [16,64,256]
    const float* keys    = (const float*)d_in[1];  // [16,1024,256]
    const float* values  = (const float*)d_in[2];  // [16,1024,256]
    const int*   vlens   = (const int*)  d_in[3];  // [16]
    const float* W_q     = (const float*)d_in[4];  // [256,128]
    const float* W_k     = (const float*)d_in[5];  // [256,128]
    const float* w_v     = (const float*)d_in[6];  // [128]
    float* out = (float*)d_out;                    // [16,64,256]

    float* ws    = (float*)d_ws;
    float* qproj = ws;                       // 1024  x 128
    float* kproj = ws + 131072;              // 16384 x 128
    float* probs = ws + 131072 + 2097152;    // 1024  x 1024

    // q projection: M=1024, K=256, N=128 -> 64*8 = 512 tiles -> 64 blocks x 8 waves
    wmma_gemm_f32<<<64, 256, 0, stream>>>(queries, W_q, qproj,
                                          256, 128, 512, 8, 0, 0, 0);
    // k projection: M=16384, K=256, N=128 -> 1024*8 = 8192 tiles -> 1024 blocks
    wmma_gemm_f32<<<1024, 256, 0, stream>>>(keys, W_k, kproj,
                                            256, 128, 8192, 8, 0, 0, 0);
    // scores + masked softmax: one block per (b,q) row
    score_softmax_kernel<<<1024, 256, 0, stream>>>(qproj, kproj, w_v, vlens, probs);
    // out[b] = probs[b] (64x1024) @ values[b] (1024x256): 16 batches * 64 tiles
    wmma_gemm_f32<<<128, 256, 0, stream>>>(probs, values, out,
                                           1024, 256, 64, 16,
                                           64L * 1024L, 1024L * 256L, 64L * 256L);
}